// RCE_25855703122464
// MI455X (gfx1250) — compile-verified
//
#include <hip/hip_runtime.h>
#include <cstdint>

// ---------------------------------------------------------------------------
// Reverse cross-entropy loss, MI455X (gfx1250).
//   entropy[b] = k * ( sum_c in[b,c] + (C-1)*LSE_b - in[b, argmin(tgt[b,:])] )
//   LSE_b      = ln sum_c exp(-in[b,c])   (computed online in base-2)
//   loss       = mean_b entropy[b],  k = RATIO/(C-1)
// Memory-bound: 1.05 GB streamed once at 23.3 TB/s (~45 us floor) -> async
// global->LDS double buffering (ASYNCcnt) is the CDNA5 data-movement path.
// ---------------------------------------------------------------------------

constexpr int   kC       = 32000;
constexpr int   kThreads = 320;               // 10 wave32s
constexpr int   kTile    = kThreads * 4;      // 1280 floats / array / tile
constexpr int   kNTiles  = kC / kTile;        // 25, exact (no tail)
constexpr float kScale   = 1.0f / (float)(kC - 1);   // RATIO = 1.0
constexpr float kLog2e   = 1.4426950408889634f;
constexpr float kLn2     = 0.6931471805599453f;

// Types matching the builtin's signature: (v4i addrspace(1)*, v4i addrspace(3)*, Ii, Ii)
typedef int v4i __attribute__((vector_size(16)));
typedef __attribute__((address_space(1))) v4i* as1_v4i_p;
typedef __attribute__((address_space(3))) v4i* as3_v4i_p;

// --- CDNA5 async global->LDS copy (16B per lane), ASYNCcnt-tracked ----------
// cpol = 1 (TH_NT): stream is read exactly once; don't thrash caches.
__device__ __forceinline__ void async_load_b128(const void* gsrc, void* lds_dst) {
#if __has_builtin(__builtin_amdgcn_global_load_async_to_lds_b128)
  __builtin_amdgcn_global_load_async_to_lds_b128(
      (as1_v4i_p)(uintptr_t)gsrc,
      (as3_v4i_p)(unsigned int)(uintptr_t)lds_dst,
      0 /*imm offset*/, 1 /*cpol: TH_NT*/);
#else
  const unsigned int laddr = (unsigned int)(uintptr_t)lds_dst;
  asm volatile("global_load_async_to_lds_b128 %0, %1, off th:TH_LOAD_NT"
               :: "v"(laddr), "v"(gsrc) : "memory");
#endif
}

template <int N>
__device__ __forceinline__ void wait_asynccnt_le() {
#if __has_builtin(__builtin_amdgcn_s_wait_asynccnt)
  __builtin_amdgcn_s_wait_asynccnt(N);
#else
  asm volatile("s_wait_asynccnt %0" :: "n"(N) : "memory");
#endif
  asm volatile("" ::: "memory");   // keep LDS reads below the wait
}

// --- merge helper for online base-2 log-sum-exp -----------------------------
__device__ __forceinline__ void lse_merge(float& m, float& s, float om, float os) {
  const float nm = fmaxf(m, om);
  s = s * __builtin_amdgcn_exp2f(m - nm) + os * __builtin_amdgcn_exp2f(om - nm);
  m = nm;
}

// Per-thread accumulator state: 4 independent slots for ILP on the exp2 chain.
struct Acc {
  float mz[4], se[4], rs[4], tm[4];
  int   ti[4];
};

__device__ __forceinline__ void process_tile(const float* __restrict__ sin,
                                             const float* __restrict__ stg,
                                             int gbase, Acc& a) {
  const float4 x4 = *(const float4*)sin;
  const float4 t4 = *(const float4*)stg;
  const float xv[4] = {x4.x, x4.y, x4.z, x4.w};
  const float tv[4] = {t4.x, t4.y, t4.z, t4.w};
#pragma unroll
  for (int j = 0; j < 4; ++j) {
    const float x = xv[j];
    a.rs[j] += x;
    const float z  = -x * kLog2e;                 // z = (-x) * log2(e)
    const float nm = fmaxf(a.mz[j], z);
    a.se[j] = a.se[j] * __builtin_amdgcn_exp2f(a.mz[j] - nm) +
              __builtin_amdgcn_exp2f(z - nm);
    a.mz[j] = nm;
    if (tv[j] < a.tm[j]) { a.tm[j] = tv[j]; a.ti[j] = gbase + j; }
  }
}

// ---------------------------------------------------------------------------
// Kernel 1: one workgroup per row; streams the row through LDS with async
// double buffering (two tiles per loop iteration so buffer indices are
// compile-time constants); writes per-row entropy to row_ent[row].
// ---------------------------------------------------------------------------
__global__ void __launch_bounds__(kThreads)
rce_row_kernel(const float* __restrict__ inp, const float* __restrict__ tgt,
               float* __restrict__ row_ent) {
  __shared__ __align__(16) float s_in[2][kTile];
  __shared__ __align__(16) float s_tg[2][kTile];
  __shared__ float r_mz[kThreads], r_s[kThreads], r_sum[kThreads], r_min[kThreads];
  __shared__ int   r_idx[kThreads];
  __shared__ float w_mz[32], w_s[32], w_sum[32], w_min[32];
  __shared__ int   w_idx[32];

  const int tid   = threadIdx.x;
  const int row   = blockIdx.x;
  const int lane4 = tid * 4;
  const float* gin = inp + (size_t)row * kC;
  const float* gtg = tgt + (size_t)row * kC;

  // Fixed LDS slots for this lane (compile-time buffer selection).
  float* const in0 = &s_in[0][lane4];
  float* const in1 = &s_in[1][lane4];
  float* const tg0 = &s_tg[0][lane4];
  float* const tg1 = &s_tg[1][lane4];

  // Prime buffer 0 with tile 0 (2 async ops in flight).
  async_load_b128(gin + lane4, in0);
  async_load_b128(gtg + lane4, tg0);

  Acc a;
#pragma unroll
  for (int j = 0; j < 4; ++j) {
    a.mz[j] = -__builtin_inff(); a.se[j] = 0.0f; a.rs[j] = 0.0f;
    a.tm[j] = __builtin_inff();  a.ti[j] = 0;
  }

  // Tiles 0..23 in 12 double-iterations; even tiles live in buf0, odd in buf1.
  for (int t = 0; t < kNTiles - 1; t += 2) {
    const int g1 = (t + 1) * kTile + lane4;   // tile t+1 -> buf1
    async_load_b128(gin + g1, in1);
    async_load_b128(gtg + g1, tg1);
    wait_asynccnt_le<2>();                    // tile t (buf0) landed
    process_tile(in0, tg0, t * kTile + lane4, a);

    const int g2 = (t + 2) * kTile + lane4;   // tile t+2 -> buf0 (t+2 <= 24)
    async_load_b128(gin + g2, in0);
    async_load_b128(gtg + g2, tg0);
    wait_asynccnt_le<2>();                    // tile t+1 (buf1) landed
    process_tile(in1, tg1, (t + 1) * kTile + lane4, a);
  }
  // Peeled final tile (kNTiles-1 = 24, even -> buf0).
  wait_asynccnt_le<0>();
  process_tile(in0, tg0, (kNTiles - 1) * kTile + lane4, a);

  // Collapse the 4 slots (prefer lower index on exact target ties).
  float cmz = a.mz[0], cse = a.se[0], csum = a.rs[0], cmin = a.tm[0];
  int   cidx = a.ti[0];
#pragma unroll
  for (int j = 1; j < 4; ++j) {
    lse_merge(cmz, cse, a.mz[j], a.se[j]);
    csum += a.rs[j];
    if (a.tm[j] < cmin || (a.tm[j] == cmin && a.ti[j] < cidx)) {
      cmin = a.tm[j]; cidx = a.ti[j];
    }
  }

  r_mz[tid] = cmz; r_s[tid] = cse; r_sum[tid] = csum; r_min[tid] = cmin; r_idx[tid] = cidx;
  __syncthreads();

  if (tid < 32) {
    float pmz = r_mz[tid], ps = r_s[tid], psum = r_sum[tid], pmin = r_min[tid];
    int   pidx = r_idx[tid];
    for (int k = tid + 32; k < kThreads; k += 32) {
      lse_merge(pmz, ps, r_mz[k], r_s[k]);
      psum += r_sum[k];
      const float om = r_min[k]; const int oi = r_idx[k];
      if (om < pmin || (om == pmin && oi < pidx)) { pmin = om; pidx = oi; }
    }
    w_mz[tid] = pmz; w_s[tid] = ps; w_sum[tid] = psum; w_min[tid] = pmin; w_idx[tid] = pidx;
  }
  __syncthreads();

  if (tid == 0) {
    float fmz = w_mz[0], fs = w_s[0], fsum = w_sum[0], fmin = w_min[0];
    int   fidx = w_idx[0];
    for (int k = 1; k < 32; ++k) {
      lse_merge(fmz, fs, w_mz[k], w_s[k]);
      fsum += w_sum[k];
      const float om = w_min[k]; const int oi = w_idx[k];
      if (om < fmin || (om == fmin && oi < fidx)) { fmin = om; fidx = oi; }
    }
    const float lse  = kLn2 * (fmz + __builtin_amdgcn_logf(fs)); // ln of sum exp(-x)
    const float xmin = gin[fidx];                                // input at argmin(target)
    row_ent[row] = kScale * (fsum + (float)(kC - 1) * lse - xmin);
  }
}

// ---------------------------------------------------------------------------
// Kernel 2: deterministic mean of B per-row values (no float atomics).
// ---------------------------------------------------------------------------
__global__ void __launch_bounds__(256)
mean_kernel(const float* __restrict__ v, float* __restrict__ out, int n) {
  __shared__ float red[256];
  float a = 0.0f;
  for (int i = threadIdx.x; i < n; i += 256) a += v[i];
  red[threadIdx.x] = a;
  __syncthreads();
  for (int s = 128; s > 0; s >>= 1) {
    if (threadIdx.x < s) red[threadIdx.x] += red[threadIdx.x + s];
    __syncthreads();
  }
  if (threadIdx.x == 0) out[0] = red[0] / (float)n;
}

// ---------------------------------------------------------------------------
extern "C" void kernel_launch(void* const* d_in, const int* in_sizes, int n_in,
                              void* d_out, int out_size, void* d_ws, size_t ws_size,
                              hipStream_t stream) {
  (void)n_in; (void)out_size; (void)ws_size;
  const float* inp = (const float*)d_in[0];
  const float* tgt = (const float*)d_in[1];
  float* out = (float*)d_out;
  float* ws  = (float*)d_ws;                 // B floats of scratch (16 KB)
  const int B = in_sizes[0] / kC;            // 4096

  rce_row_kernel<<<B, kThreads, 0, stream>>>(inp, tgt, ws);
  mean_kernel<<<1, 256, 0, stream>>>(ws, out, B);
}